// HeteroGIN_24137716203677
// MI455X (gfx1250) — compile-verified
//
#include <hip/hip_runtime.h>

typedef __attribute__((ext_vector_type(2))) float v2f;
typedef __attribute__((ext_vector_type(8))) float v8f;

constexpr int   HD      = 128;
constexpr float BN_EPSF = 1e-5f;

// ---------------------------------------------------------------- zero fill
__global__ void k_zero(float* __restrict__ p, long long n) {
  long long i  = (long long)blockIdx.x * blockDim.x + threadIdx.x;
  long long st = (long long)gridDim.x * blockDim.x;
  for (; i < n; i += st) p[i] = 0.0f;
}

// ---------------------------------------------------------- edge scatter-add
// 32 lanes per edge: each lane gathers a float4 of the source row (contiguous
// 512B -> fully coalesced, L2 resident) and atomically adds into dst row.
__global__ void k_scatter_add(const float* __restrict__ h,
                              const int* __restrict__ src,
                              const int* __restrict__ dst,
                              float* __restrict__ agg, int E) {
  long long t     = (long long)blockIdx.x * blockDim.x + threadIdx.x;
  long long total = (long long)E * (HD / 4);
  if (t >= total) return;
  int e = (int)(t >> 5);      // HD/4 == 32 quads per edge
  int q = (int)(t & 31);
  int s = src[e];
  int d = dst[e];
  const float4 v = *reinterpret_cast<const float4*>(h + (long long)s * HD + q * 4);
  float* o = agg + (long long)d * HD + q * 4;
  __hip_atomic_fetch_add(o + 0, v.x, __ATOMIC_RELAXED, __HIP_MEMORY_SCOPE_AGENT);
  __hip_atomic_fetch_add(o + 1, v.y, __ATOMIC_RELAXED, __HIP_MEMORY_SCOPE_AGENT);
  __hip_atomic_fetch_add(o + 2, v.z, __ATOMIC_RELAXED, __HIP_MEMORY_SCOPE_AGENT);
  __hip_atomic_fetch_add(o + 3, v.w, __ATOMIC_RELAXED, __HIP_MEMORY_SCOPE_AGENT);
}

// ------------------------------------------------- per-column sum / sumsq
__global__ void k_colstats(const float* __restrict__ x, int M,
                           float* __restrict__ stats /* [2*HD] */) {
  const int col = threadIdx.x;          // blockDim.x == HD
  float s = 0.0f, ss = 0.0f;
  for (int r = blockIdx.x; r < M; r += gridDim.x) {
    float v = x[(long long)r * HD + col];
    s  += v;
    ss += v * v;
  }
  __hip_atomic_fetch_add(&stats[col],      s,  __ATOMIC_RELAXED, __HIP_MEMORY_SCOPE_AGENT);
  __hip_atomic_fetch_add(&stats[HD + col], ss, __ATOMIC_RELAXED, __HIP_MEMORY_SCOPE_AGENT);
}

__global__ void k_bn_finalize(const float* __restrict__ stats,
                              const float* __restrict__ g, float Minv,
                              float* __restrict__ mu, float* __restrict__ rsg) {
  int c = threadIdx.x;                  // blockDim.x == HD
  float m   = stats[c] * Minv;
  float var = stats[HD + c] * Minv - m * m;   // biased variance (matches ref)
  mu[c]  = m;
  rsg[c] = g[c] * rsqrtf(var + BN_EPSF);
}

// ----------------------------------------------------------- WMMA fp32 GEMM
// C[M,N] = f(A)[M,K] @ B[K,N] + bias. Each wave owns one 16-wide N tile and
// MT stacked 16-high M tiles: one B fragment feeds MT independent WMMAs.
// MODE 0: f(A)=A
// MODE 1: f(A) = (1+eps)*A + AGG        (p0=AGG, p1=eps ptr)      [GIN combine]
// MODE 2: f(A) = relu((A-mu[k])*rsg[k]+beta[k]) (p0=mu,p1=rsg,p2=beta) [BN+ReLU]
// A 16x4 layout : lane%16 = M row; lane/16 selects K pair {0,1}/{2,3}
// B 4x16 layout : lane%16 = N col; same K-pair split
// C/D layout    : VGPR r -> M = r + 8*(lane/16), N = lane%16
template <int MODE, int MT, bool NMASK>
__global__ void __launch_bounds__(128)
k_wmma_gemm(const float* __restrict__ A, const float* __restrict__ B,
            const float* __restrict__ bias, float* __restrict__ C,
            int M, int K, int N,
            const float* __restrict__ p0, const float* __restrict__ p1,
            const float* __restrict__ p2, int relu_out) {
  const int lane  = threadIdx.x & 31;
  const int wave  = threadIdx.x >> 5;
  const int tileN = blockIdx.x * 4 + wave;
  if (tileN * 16 >= N) return;                       // wave-uniform exit

  const int  ncol  = tileN * 16 + (lane & 15);
  const bool colOK = NMASK ? (ncol < N) : true;
  const int  ncolc = NMASK ? (colOK ? ncol : (N - 1)) : ncol;  // clamped, in-bounds
  const int  koff  = (lane >> 4) * 2;

  const int m0 = blockIdx.y * MT * 16 + (lane & 15); // first M tile's row
  const float* Ar[MT];
  const float* Gr[MT];
#pragma unroll
  for (int i = 0; i < MT; ++i) {
    Ar[i] = A + (long long)(m0 + i * 16) * K;
    Gr[i] = (MODE == 1) ? (p0 + (long long)(m0 + i * 16) * K) : nullptr;
  }
  const float  epsf = (MODE == 1) ? (1.0f + p1[0]) : 0.0f;
  const float* Bp   = B + ncolc;

  const v8f vzero = {};
  v8f acc[MT];
#pragma unroll
  for (int i = 0; i < MT; ++i) acc[i] = vzero;

  for (int k = 0; k < K; k += 4) {
    const int ka = k + koff;
    // B fragment: unconditional loads (address clamped), mask via select
    v2f bv;
    {
      float bx = Bp[(long long)ka * N];
      float by = Bp[(long long)(ka + 1) * N];
      bv.x = (!NMASK || colOK) ? bx : 0.0f;
      bv.y = (!NMASK || colOK) ? by : 0.0f;
    }
    float s0 = 0.0f, s1 = 0.0f, g0 = 0.0f, g1 = 0.0f;   // MODE-2 per-K params
    if (MODE == 2) {
      s0 = p0[ka]; s1 = p0[ka + 1];
      g0 = p1[ka]; g1 = p1[ka + 1];
    }
#pragma unroll
    for (int i = 0; i < MT; ++i) {
      float a0 = Ar[i][ka];
      float a1 = Ar[i][ka + 1];
      if (MODE == 1) {
        a0 = fmaf(epsf, a0, Gr[i][ka]);
        a1 = fmaf(epsf, a1, Gr[i][ka + 1]);
      } else if (MODE == 2) {
        a0 = fmaxf(fmaf(a0 - s0, g0, p2[ka]),     0.0f);
        a1 = fmaxf(fmaf(a1 - s1, g1, p2[ka + 1]), 0.0f);
      }
      v2f av; av.x = a0; av.y = a1;
      acc[i] = __builtin_amdgcn_wmma_f32_16x16x4_f32(
          /*neg_a=*/false, av, /*neg_b=*/false, bv,
          /*c_mod=*/(short)0, acc[i], /*reuse_a=*/false, /*reuse_b=*/false);
    }
  }

  if (colOK) {
    const float bb = bias ? bias[ncol] : 0.0f;
#pragma unroll
    for (int i = 0; i < MT; ++i) {
      const int rbase = (blockIdx.y * MT + i) * 16 + ((lane >> 4) << 3);
#pragma unroll
      for (int r = 0; r < 8; ++r) {
        float v = acc[i][r] + bb;
        if (relu_out) v = fmaxf(v, 0.0f);
        C[(long long)(rbase + r) * N + ncol] = v;
      }
    }
  }
}

template <int MODE>
static void launch_gemm(const float* A, const float* B, const float* bias, float* C,
                        int M, int K, int N,
                        const float* p0, const float* p1, const float* p2,
                        int relu_out, hipStream_t s) {
  const int  tilesN = (N + 15) / 16;
  const bool nmask  = (N % 16) != 0;
  const dim3 blk(128);
  if (M % (16 * 5) == 0) {
    dim3 grid((tilesN + 3) / 4, M / (16 * 5));
    if (nmask) k_wmma_gemm<MODE, 5, true ><<<grid, blk, 0, s>>>(A, B, bias, C, M, K, N, p0, p1, p2, relu_out);
    else       k_wmma_gemm<MODE, 5, false><<<grid, blk, 0, s>>>(A, B, bias, C, M, K, N, p0, p1, p2, relu_out);
  } else {
    dim3 grid((tilesN + 3) / 4, M / 16);
    if (nmask) k_wmma_gemm<MODE, 1, true ><<<grid, blk, 0, s>>>(A, B, bias, C, M, K, N, p0, p1, p2, relu_out);
    else       k_wmma_gemm<MODE, 1, false><<<grid, blk, 0, s>>>(A, B, bias, C, M, K, N, p0, p1, p2, relu_out);
  }
}

// ---------------------------------------------------------------------------
extern "C" void kernel_launch(void* const* d_in, const int* in_sizes, int n_in,
                              void* d_out, int out_size, void* d_ws, size_t ws_size,
                              hipStream_t stream) {
  (void)n_in; (void)out_size; (void)ws_size;

  const float* x_author = (const float*)d_in[0];
  const float* x_paper  = (const float*)d_in[1];
  const int*   w_src    = (const int*)d_in[2];
  const int*   w_dst    = (const int*)d_in[3];
  const int*   wbsrc    = (const int*)d_in[4];
  const int*   wbdst    = (const int*)d_in[5];
  const float* la_w = (const float*)d_in[6];
  const float* la_b = (const float*)d_in[7];
  const float* lp_w = (const float*)d_in[8];
  const float* lp_b = (const float*)d_in[9];
  const float* eps  = (const float*)d_in[10];
  const float* w1   = (const float*)d_in[11];
  const float* b1   = (const float*)d_in[12];
  const float* bng  = (const float*)d_in[13];
  const float* bnb  = (const float*)d_in[14];
  const float* w2   = (const float*)d_in[15];
  const float* b2   = (const float*)d_in[16];
  const float* fw   = (const float*)d_in[17];
  const float* fb   = (const float*)d_in[18];

  const int NA = in_sizes[0] / 128;   // 50000
  const int NP = in_sizes[1] / 256;   // 100000
  const int E  = in_sizes[2];         // 1000000
  const int NC = in_sizes[18];        // 40
  const int L  = in_sizes[10] / 2;    // eps is [L,2]

  // workspace layout (floats)
  float* ws = (float*)d_ws;
  size_t off = 0;
  float* h_a   = ws + off; off += (size_t)NA * HD;
  float* h_p   = ws + off; off += (size_t)NP * HD;
  float* agg_a = ws + off; off += (size_t)NA * HD;
  float* agg_p = ws + off; off += (size_t)NP * HD;
  float* tmp   = ws + off; off += (size_t)NP * HD;  // MLP hidden (pre-BN)
  float* stats = ws + off; off += 2 * HD;
  float* mu    = ws + off; off += HD;
  float* rsg   = ws + off; off += HD;

  // input projections (fp32 WMMA)
  launch_gemm<0>(x_author, la_w, la_b, h_a, NA, 128, HD, nullptr, nullptr, nullptr, 0, stream);
  launch_gemm<0>(x_paper,  lp_w, lp_b, h_p, NP, 256, HD, nullptr, nullptr, nullptr, 0, stream);

  const long long tS = (long long)E * (HD / 4);
  const int gS = (int)((tS + 255) / 256);

  for (int l = 0; l < L; ++l) {
    // zero aggregation buffers, then both scatter-sums on the OLD h
    k_zero<<<2048, 256, 0, stream>>>(agg_p, (long long)NP * HD);
    k_zero<<<2048, 256, 0, stream>>>(agg_a, (long long)NA * HD);
    k_scatter_add<<<gS, 256, 0, stream>>>(h_a, w_src, w_dst, agg_p, E);
    k_scatter_add<<<gS, 256, 0, stream>>>(h_p, wbsrc, wbdst, agg_a, E);

    // t = 0 : papers      t = 1 : authors
    for (int t = 0; t < 2; ++t) {
      const int    Mn  = (t == 0) ? NP : NA;
      float*       h   = (t == 0) ? h_p : h_a;
      float*       agg = (t == 0) ? agg_p : agg_a;
      const size_t pi  = (size_t)(l * 2 + t);
      const float* W1  = w1  + pi * HD * HD;
      const float* B1  = b1  + pi * HD;
      const float* G   = bng + pi * HD;
      const float* Bt  = bnb + pi * HD;
      const float* W2  = w2  + pi * HD * HD;
      const float* B2  = b2  + pi * HD;
      const float* ep  = eps + pi;

      // tmp = ((1+eps)*h + agg) @ W1 + b1      (combine fused into A path)
      launch_gemm<1>(h, W1, B1, tmp, Mn, HD, HD, agg, ep, nullptr, 0, stream);

      // batch statistics of tmp columns
      k_zero<<<1, 256, 0, stream>>>(stats, 2 * HD);
      k_colstats<<<2048, HD, 0, stream>>>(tmp, Mn, stats);
      k_bn_finalize<<<1, HD, 0, stream>>>(stats, G, 1.0f / (float)Mn, mu, rsg);

      // h = relu( bn_relu(tmp) @ W2 + b2 )     (BN+ReLU fused into A path,
      //                                         outer ReLU fused into epilogue)
      launch_gemm<2>(tmp, W2, B2, h, Mn, HD, HD, mu, rsg, Bt, 1, stream);
    }
  }

  // classifier head: out[NA, NC] = h_a @ final_w + final_b
  launch_gemm<0>(h_a, fw, fb, (float*)d_out, NA, HD, NC, nullptr, nullptr, nullptr, 0, stream);
}